// RandomCutout_59545426592097
// MI455X (gfx1250) — compile-verified
//
#include <hip/hip_runtime.h>

// RandomCutout for [B=32, H=512, W=512, C=3] f32, NUM_MASKS=2, 128x128 masks.
// Pure streaming op: ~201 MB moved, roofline ~8.6 us @ 23.3 TB/s HBM.
// Strategy: 128-bit non-temporal vector memory per lane (wave32 -> 512 B/wave
// per access), block-uniform mask parameters via scalar loads, zero divergence.

typedef __attribute__((ext_vector_type(4))) float f4;

#define B_IMG   32
#define H_IMG   512
#define W_IMG   512
#define C_IMG   3
#define ROW_ELEMS   (W_IMG * C_IMG)          // 1536 floats per row (mult of 4)
#define IMG_ELEMS   (H_IMG * ROW_ELEMS)      // 786432 floats per image
#define VEC_PER_IMG (IMG_ELEMS / 4)          // 196608 float4 per image
#define BLK 256
#define BLOCKS_PER_IMG (VEC_PER_IMG / BLK)   // 768 blocks per image (exact)
#define HALF_MASK 64                          // MASK/2

__global__ __launch_bounds__(BLK) void RandomCutout_59545426592097_kernel(
    const float* __restrict__ in,
    const int*   __restrict__ centers_y,   // [B, 2]
    const int*   __restrict__ centers_x,   // [B, 2]
    float*       __restrict__ out)
{
    // Block-uniform image index: blocks never straddle an image because
    // IMG_ELEMS is a multiple of (BLK*4).
    const int b = (int)blockIdx.x / BLOCKS_PER_IMG;      // uniform -> SGPR

    // float4 index within the whole tensor.
    const int g  = (int)blockIdx.x * BLK + (int)threadIdx.x;
    // element offset within this image (multiple of 4)
    const int eb = (g - b * VEC_PER_IMG) * 4;

    // eb -> (y, x0, c0). Constant-divisor divides lower to mul+shift.
    const int y  = eb / ROW_ELEMS;
    const int r  = eb - y * ROW_ELEMS;       // 0..1532, multiple of 4
    const int x0 = r / C_IMG;
    const int c0 = r - x0 * C_IMG;           // 0..2

    // Mask parameters: uniform per block -> scalar loads (s_load_b64).
    const int cy0 = centers_y[b * 2 + 0];
    const int cy1 = centers_y[b * 2 + 1];
    const int cx0 = centers_x[b * 2 + 0];
    const int cx1 = centers_x[b * 2 + 1];

    // Reference clips to [0,H]/[0,W]; since y,x already lie in-range the raw
    // compares are exact (signed compares handle cy-64 < 0 correctly).
    const bool iny0 = (y >= cy0 - HALF_MASK) & (y < cy0 + HALF_MASK);
    const bool iny1 = (y >= cy1 - HALF_MASK) & (y < cy1 + HALF_MASK);

    const int x1 = x0 + 1;                   // the only other column this f4 touches
    const bool cutA = (iny0 & (x0 >= cx0 - HALF_MASK) & (x0 < cx0 + HALF_MASK)) |
                      (iny1 & (x0 >= cx1 - HALF_MASK) & (x0 < cx1 + HALF_MASK));
    const bool cutB = (iny0 & (x1 >= cx0 - HALF_MASK) & (x1 < cx0 + HALF_MASK)) |
                      (iny1 & (x1 >= cx1 - HALF_MASK) & (x1 < cx1 + HALF_MASK));

    // Multiply by 0/1 float exactly like the reference (keep = !cut).
    const float fA = cutA ? 0.0f : 1.0f;
    const float fB = cutB ? 0.0f : 1.0f;

    // Streaming 128-bit access, non-temporal (read-once / write-once data):
    // global_load_b128 ... th:TH_LOAD_NT / global_store_b128 ... th:TH_STORE_NT
    f4 v = __builtin_nontemporal_load((const f4*)in + g);

    // Channel -> column mapping inside the float4:
    //   lane 0: column x0 always;  lane 3: column x0+1 always;
    //   lane 1: x0 iff c0 < 2;     lane 2: x0 iff c0 == 0.
    v.x *= fA;
    v.y *= (c0 < 2)  ? fA : fB;
    v.z *= (c0 == 0) ? fA : fB;
    v.w *= fB;

    __builtin_nontemporal_store(v, (f4*)out + g);
}

extern "C" void kernel_launch(void* const* d_in, const int* in_sizes, int n_in,
                              void* d_out, int out_size, void* d_ws, size_t ws_size,
                              hipStream_t stream) {
    const float* in  = (const float*)d_in[0];   // [32,512,512,3] f32
    const int* cy    = (const int*)d_in[1];     // [32,2] i32
    const int* cx    = (const int*)d_in[2];     // [32,2] i32
    float* out       = (float*)d_out;           // [32,512,512,3] f32

    const int grid = B_IMG * BLOCKS_PER_IMG;    // 24576 blocks of 256 (8 waves)
    RandomCutout_59545426592097_kernel<<<grid, BLK, 0, stream>>>(in, cy, cx, out);
}